// InterpLnr_73675868996021
// MI455X (gfx1250) — compile-verified
//
#include <hip/hip_runtime.h>

typedef float v4f __attribute__((ext_vector_type(4)));

#define T_PAD   192   // MAX_LEN_PAD
#define CCH     256   // C
#define NSEG    7     // MAX_NUM_SEG
#define L2SEG   64    // 2 * MAX_LEN_SEG
#define MAXROWS 192   // >= max_ilen (<=128) with margin
#define BLOCK   512   // 16 wave32 waves

__global__ __launch_bounds__(BLOCK) void interp_lnr_kernel(
    const float* __restrict__ x,
    const float* __restrict__ scales,
    const int*   __restrict__ len_seq,
    const int*   __restrict__ len_seg,
    float*       __restrict__ out,
    int max_ilen)
{
    __shared__ int   s_row[MAXROWS];
    __shared__ float s_lam[MAXROWS];
    __shared__ int   s_start[NSEG];
    __shared__ int   s_n[NSEG];
    __shared__ int   s_off[NSEG];
    __shared__ int   s_total;

    const int b   = blockIdx.x;
    const int tid = threadIdx.x;

    // ---- Phase A: per-segment valid counts + prefix (wave 0 only) ----
    if (tid < 32) {
        const int lane = tid;
        const int lseq = len_seq[b];
        int start = 0;
        int off   = 0;
        #pragma unroll
        for (int s = 0; s < NSEG; ++s) {
            const int   seg = len_seg[b * NSEG + s];
            const float sc  = scales[b * NSEG + s];
            int lim  = seg - 1;
            const int lim2 = lseq - 1 - off;
            if (lim2 < lim) lim = lim2;
            int n = 0;
            if (lim > 0) {
                // exact per-j evaluation of floor(j/scale) < lim, j in [0,64)
                const float fl0 = floorf((float)lane        / sc);
                const float fl1 = floorf((float)(lane + 32) / sc);
                const unsigned m0 = __builtin_amdgcn_ballot_w32(fl0 < (float)lim);
                const unsigned m1 = __builtin_amdgcn_ballot_w32(fl1 < (float)lim);
                n = __popc(m0) + __popc(m1);   // valid j's form a contiguous prefix
            }
            if (lane == 0) { s_start[s] = start; s_n[s] = n; s_off[s] = off; }
            start += n;
            off   += seg;
        }
        if (lane == 0) s_total = (start < max_ilen) ? start : max_ilen;
    }
    __syncthreads();

    // ---- Phase A2: build compact row table (pos -> source row, lambda) ----
    if (tid < NSEG * L2SEG) {
        const int s = tid >> 6;
        const int j = tid & 63;
        if (j < s_n[s]) {
            const int p = s_start[s] + j;
            if (p < max_ilen) {
                const float sc = scales[b * NSEG + s];
                const float q  = (float)j / sc;
                const float fl = floorf(q);
                s_row[p] = (int)fl + s_off[s];
                s_lam[p] = q - fl;
            }
        }
    }
    __syncthreads();

    // ---- Phase B: stream rows; one wave per output row, 8 floats/lane ----
    const int nwaves = BLOCK / 32;
    const int wid    = tid >> 5;
    const int lane   = tid & 31;
    const int c0     = lane * 8;
    const int total  = s_total;
    const float* __restrict__ xb = x   + (size_t)b * (size_t)(T_PAD * CCH);
    float*       __restrict__ ob = out + (size_t)b * (size_t)max_ilen * CCH;

    // Software pipeline: keep the next row-pair's loads in flight while
    // finishing the current one (double the per-wave bytes outstanding).
    v4f A0 = (v4f)(0.0f), A1 = (v4f)(0.0f), B0 = (v4f)(0.0f), B1 = (v4f)(0.0f);
    float lam_c = 0.0f;

    // prologue: issue loads for the first row owned by this wave
    if (wid < total) {
        const int   r  = s_row[wid];
        lam_c          = s_lam[wid];
        const float* pa = xb + (size_t)r * CCH + c0;
        A0 = *(const v4f*)(pa);
        A1 = *(const v4f*)(pa + 4);
        B0 = *(const v4f*)(pa + CCH);
        B1 = *(const v4f*)(pa + CCH + 4);
    }

    for (int p = wid; p < max_ilen; p += nwaves) {
        const int pn = p + nwaves;

        // prefetch two iterations ahead (rows overlap across p -> keep resident)
        const int pf = pn + nwaves;
        if (pf < total) {
            const int rf = s_row[pf];
            __builtin_prefetch(xb + (size_t)rf * CCH + c0,       0, 1);
            __builtin_prefetch(xb + (size_t)(rf + 1) * CCH + c0, 0, 1);
        }

        // issue next iteration's gather loads before consuming current data
        v4f NA0 = (v4f)(0.0f), NA1 = (v4f)(0.0f);
        v4f NB0 = (v4f)(0.0f), NB1 = (v4f)(0.0f);
        float lam_n = 0.0f;
        if (pn < total) {
            const int   rn  = s_row[pn];
            lam_n           = s_lam[pn];
            const float* pb = xb + (size_t)rn * CCH + c0;
            NA0 = *(const v4f*)(pb);
            NA1 = *(const v4f*)(pb + 4);
            NB0 = *(const v4f*)(pb + CCH);
            NB1 = *(const v4f*)(pb + CCH + 4);
        }

        // consume current row (zeros if in the padded tail)
        v4f r0, r1;
        if (p < total) {
            const float w = 1.0f - lam_c;
            r0 = w * A0 + lam_c * B0;
            r1 = w * A1 + lam_c * B1;
        } else {
            r0 = (v4f)(0.0f);
            r1 = (v4f)(0.0f);
        }
        // output is write-once: non-temporal stores, don't pollute caches
        __builtin_nontemporal_store(r0, (v4f*)(ob + (size_t)p * CCH + c0));
        __builtin_nontemporal_store(r1, (v4f*)(ob + (size_t)p * CCH + c0 + 4));

        // rotate pipeline registers
        A0 = NA0; A1 = NA1; B0 = NB0; B1 = NB1; lam_c = lam_n;
    }
}

extern "C" void kernel_launch(void* const* d_in, const int* in_sizes, int n_in,
                              void* d_out, int out_size, void* d_ws, size_t ws_size,
                              hipStream_t stream) {
    const float* x       = (const float*)d_in[0];
    const float* scales  = (const float*)d_in[1];
    const int*   len_seq = (const int*)d_in[2];
    const int*   len_seg = (const int*)d_in[3];
    float*       out     = (float*)d_out;

    const int B        = in_sizes[2];               // len_seq has B elements
    const int max_ilen = out_size / (B * CCH);      // = max(len_seq) in reference

    interp_lnr_kernel<<<B, BLOCK, 0, stream>>>(x, scales, len_seq, len_seg, out, max_ilen);
}